// DeepJDOT_loss_21122649162604
// MI455X (gfx1250) — compile-verified
//
#include <hip/hip_runtime.h>
#include <hip/hip_bf16.h>
#include <math.h>

typedef __attribute__((ext_vector_type(2))) float v2f;
typedef __attribute__((ext_vector_type(8))) float v8f;

#define BS 64
#define BT 64
#define TS 128
#define TT 128
#define QD 64
#define ALPHA 1.0f
#define BETA 0.1f

// ---------------------------------------------------------------------------
// Kernel 0: row squared norms of xs (BS*TS rows) and xt (BT*TT rows), Q=64
// ---------------------------------------------------------------------------
__global__ __launch_bounds__(256) void norms_kernel(
    const float* __restrict__ xs, const float* __restrict__ xt,
    float* __restrict__ x2, float* __restrict__ y2) {
  int idx = blockIdx.x * 256 + threadIdx.x;  // 0 .. 16383
  const float* src;
  float* dst;
  if (idx < BS * TS) {
    src = xs + (size_t)idx * QD;
    dst = x2 + idx;
  } else {
    int k = idx - BS * TS;
    src = xt + (size_t)k * QD;
    dst = y2 + k;
  }
  float s = 0.f;
#pragma unroll 8
  for (int q = 0; q < QD; ++q) {
    float v = src[q];
    s += v * v;
  }
  *dst = s;
}

// ---------------------------------------------------------------------------
// Kernel 1: one block per (s,t) pair.
//   Phase 1: WMMA f32 16x16x4 -> 128x128 dot products -> L2 distances in LDS
//   Phase 2: DTW cumulative-cost DP, anti-diagonal wavefront, in-place in LDS
//   Phase 3: single-thread greedy backtrack (numpy argmin tie order) -> cost
// ---------------------------------------------------------------------------
__global__ __launch_bounds__(256) void dtw_pair_kernel(
    const float* __restrict__ xs, const float* __restrict__ xt,
    const float* __restrict__ x2g, const float* __restrict__ y2g,
    const float* __restrict__ simCE,
    float* __restrict__ dtw, float* __restrict__ costOT) {
  __shared__ float Cm[TS * TT];  // 64 KB: dists, then overwritten by acc

  const int p = blockIdx.x;     // pair index
  const int s = p >> 6;
  const int t = p & 63;
  const int tid = threadIdx.x;
  const int wave = tid >> 5;    // 0..7  -> tile row
  const int lane = tid & 31;

  const float* As = xs + (size_t)s * TS * QD;
  const float* Bt = xt + (size_t)t * TT * QD;

  // WMMA fragment addressing (16x16x4 f32):
  //   A: lane l holds A[m][k], m = l&15, k = (l>>4)*2 + v  (v = vgpr 0/1)
  //   B: lane l holds B[k][n], n = l&15, k = (l>>4)*2 + v
  const int mrow = lane & 15;
  const int khalf = (lane >> 4) << 1;  // 0 or 2
  const int i0 = wave * 16;

  // Hoist the 16 A fragments (K = 64 = 16 steps of 4) for this tile row.
  v2f afrag[16];
#pragma unroll
  for (int kb = 0; kb < 16; ++kb) {
    afrag[kb] = *(const v2f*)(As + (i0 + mrow) * QD + kb * 4 + khalf);
  }

  const int mbase = (lane >> 4) * 8;
  const int ncol = lane & 15;

  for (int tj = 0; tj < 8; ++tj) {
    const int j0 = tj * 16;
    v8f acc = {};
#pragma unroll
    for (int kb = 0; kb < 16; ++kb) {
      v2f b = *(const v2f*)(Bt + (j0 + mrow) * QD + kb * 4 + khalf);
      acc = __builtin_amdgcn_wmma_f32_16x16x4_f32(
          false, afrag[kb], false, b, (short)0, acc, false, false);
    }
    // D layout: vgpr r, lanes 0-15 -> (m=r, n=lane); lanes 16-31 -> (m=8+r)
    const float y2v = y2g[t * TT + j0 + ncol];
#pragma unroll
    for (int r = 0; r < 8; ++r) {
      const int i = i0 + mbase + r;
      const float x2v = x2g[s * TS + i];
      const float d2 = x2v + y2v - 2.0f * acc[r];
      Cm[i * TT + (j0 + ncol)] = sqrtf(fmaxf(d2, 1e-12f));
    }
  }
  __syncthreads();

  // ---- Phase 2: wavefront DP, in place. Thread j owns column j. ----
  const float INF = __builtin_inff();
  const int j = tid;
  for (int d = 0; d < TS + TT - 1; ++d) {
    if (j < TT) {
      const int i = d - j;
      if (i >= 0 && i < TS) {
        const float cv = Cm[i * TT + j];
        const float up = (i > 0) ? Cm[(i - 1) * TT + j] : INF;
        const float lf = (j > 0) ? Cm[i * TT + (j - 1)] : INF;
        const float dg = (i > 0 && j > 0) ? Cm[(i - 1) * TT + (j - 1)] : INF;
        float m = fminf(up, fminf(lf, dg));
        if (d == 0) m = 0.0f;  // acc[0][0] = C[0][0]
        Cm[i * TT + j] = cv + m;
      }
    }
    __syncthreads();
  }

  // ---- Phase 3: backtrack. dist(cell) = acc - min(predecessors). ----
  if (tid == 0) {
    float cost = 0.0f;
    int bi = TS - 1, bj = TT - 1;  // end cell is NOT marked
    for (int step = 0; step < TS + TT - 2; ++step) {
      if (bi == 0 && bj == 0) break;
      int ni, nj;
      if (bi == 0) {
        ni = 0; nj = bj - 1;
      } else if (bj == 0) {
        ni = bi - 1; nj = 0;
      } else {
        const float cd = Cm[(bi - 1) * TT + (bj - 1)];
        const float cu = Cm[(bi - 1) * TT + bj];
        const float cl = Cm[bi * TT + (bj - 1)];
        int am = 0;
        float best = cd;
        if (cu < best) { best = cu; am = 1; }
        if (cl < best) { best = cl; am = 2; }
        ni = (am == 2) ? bi : (bi - 1);
        nj = (am == 1) ? bj : (bj - 1);
      }
      float dist;
      if (ni == 0 && nj == 0) {
        dist = Cm[0];
      } else if (ni == 0) {
        dist = Cm[nj] - Cm[nj - 1];
      } else if (nj == 0) {
        dist = Cm[ni * TT] - Cm[(ni - 1) * TT];
      } else {
        const float pd = Cm[(ni - 1) * TT + (nj - 1)];
        const float pu = Cm[(ni - 1) * TT + nj];
        const float pl = Cm[ni * TT + (nj - 1)];
        dist = Cm[ni * TT + nj] - fminf(pd, fminf(pu, pl));
      }
      cost += dist;
      bi = ni;
      bj = nj;
    }
    dtw[p] = cost;
    costOT[p] = ALPHA * cost + BETA * simCE[p];
  }
}

// ---------------------------------------------------------------------------
// Kernel 2: single block. eps = 0.02*max(M); 300 log-domain Sinkhorn iters;
// gamma = exp((f+g-M)/eps); out = [A*sum(g*dtw)/64, B*sum(g*sim)/64, gamma]
// ---------------------------------------------------------------------------
__global__ __launch_bounds__(256) void sinkhorn_kernel(
    const float* __restrict__ costOT, const float* __restrict__ dtw,
    const float* __restrict__ simCE, float* __restrict__ out) {
  __shared__ float M[BS * BT];
  __shared__ float f[BS];
  __shared__ float g[BT];
  __shared__ float red[256];

  const int tid = threadIdx.x;
  const float NEG_INF = -__builtin_inff();

  float mx = NEG_INF;
  for (int k = tid; k < BS * BT; k += 256) {
    const float v = costOT[k];
    M[k] = v;
    mx = fmaxf(mx, v);
  }
  red[tid] = mx;
  __syncthreads();
  for (int st = 128; st > 0; st >>= 1) {
    if (tid < st) red[tid] = fmaxf(red[tid], red[tid + st]);
    __syncthreads();
  }
  const float eps = 0.02f * red[0];
  const float inv_eps = 1.0f / eps;
  const float loga = -logf(64.0f);  // log(1/B_S) == log(1/B_T)

  if (tid < BS) f[tid] = 0.0f;
  if (tid < BT) g[tid] = 0.0f;
  __syncthreads();

  for (int it = 0; it < 300; ++it) {
    // f update: rows, uses g from previous iteration
    if (tid < BS) {
      const int r = tid;
      float dm = NEG_INF;
      for (int c = 0; c < BT; ++c) dm = fmaxf(dm, g[c] - M[r * BT + c]);
      float ssum = 0.0f;
      for (int c = 0; c < BT; ++c)
        ssum += expf((g[c] - M[r * BT + c] - dm) * inv_eps);
      f[r] = eps * loga - dm - eps * logf(ssum);
    }
    __syncthreads();
    // g update: columns, uses new f
    if (tid < BT) {
      const int c = tid;
      float dm = NEG_INF;
      for (int r = 0; r < BS; ++r) dm = fmaxf(dm, f[r] - M[r * BT + c]);
      float ssum = 0.0f;
      for (int r = 0; r < BS; ++r)
        ssum += expf((f[r] - M[r * BT + c] - dm) * inv_eps);
      g[c] = eps * loga - dm - eps * logf(ssum);
    }
    __syncthreads();
  }

  // gamma + weighted sums
  float pa = 0.0f, pb = 0.0f;
  for (int k = tid; k < BS * BT; k += 256) {
    const int r = k >> 6, c = k & 63;
    const float gm = expf((f[r] + g[c] - M[k]) * inv_eps);
    out[2 + k] = gm;
    pa += gm * dtw[k];
    pb += gm * simCE[k];
  }
  red[tid] = pa;
  __syncthreads();
  for (int st = 128; st > 0; st >>= 1) {
    if (tid < st) red[tid] += red[tid + st];
    __syncthreads();
  }
  const float alpha_sum = red[0];
  __syncthreads();
  red[tid] = pb;
  __syncthreads();
  for (int st = 128; st > 0; st >>= 1) {
    if (tid < st) red[tid] += red[tid + st];
    __syncthreads();
  }
  if (tid == 0) {
    const float length = (float)QD;  // (64 + 64) / 2
    out[0] = ALPHA * alpha_sum / length;
    out[1] = BETA * red[0] / length;
  }
}

// ---------------------------------------------------------------------------
extern "C" void kernel_launch(void* const* d_in, const int* in_sizes, int n_in,
                              void* d_out, int out_size, void* d_ws,
                              size_t ws_size, hipStream_t stream) {
  const float* xs = (const float*)d_in[0];     // (64,128,64) f32
  const float* xt = (const float*)d_in[1];     // (64,128,64) f32
  // d_in[2] = labels_source (unused by the reference computation)
  const float* simCE = (const float*)d_in[3];  // (64,64) f32
  float* out = (float*)d_out;                  // [a, b, gamma(4096)]

  float* x2g = (float*)d_ws;          // 8192
  float* y2g = x2g + BS * TS;         // 8192
  float* dtw = y2g + BT * TT;         // 4096
  float* costOT = dtw + BS * BT;      // 4096   (total 96 KB scratch)

  norms_kernel<<<(BS * TS + BT * TT) / 256, 256, 0, stream>>>(xs, xt, x2g, y2g);
  dtw_pair_kernel<<<BS * BT, 256, 0, stream>>>(xs, xt, x2g, y2g, simCE, dtw,
                                               costOT);
  sinkhorn_kernel<<<1, 256, 0, stream>>>(costOT, dtw, simCE, out);
}